// GraphSAGE_25159918420563
// MI455X (gfx1250) — compile-verified
//
#include <hip/hip_runtime.h>

#define NN 50000
#define NE 800000
#define F  128
#define NC 16

typedef __attribute__((ext_vector_type(2))) float v2f;
typedef __attribute__((ext_vector_type(8))) float v8f;

// ---------------------------------------------------------------- utilities
__global__ void zero_kernel(float* __restrict__ p, int n) {
    int i = blockIdx.x * blockDim.x + threadIdx.x;
    if (i < n) p[i] = 0.0f;
}

__global__ void degree_kernel(const int* __restrict__ dst, float* __restrict__ deg) {
    int e = blockIdx.x * blockDim.x + threadIdx.x;
    if (e < NE) atomicAdd(&deg[dst[e]], 1.0f);
}

__global__ void invdeg_kernel(float* __restrict__ deg) {
    int i = blockIdx.x * blockDim.x + threadIdx.x;
    if (i < NN) deg[i] = 1.0f / fmaxf(deg[i], 1.0f);
}

// ------------------------------------------------- edge gather / scatter-add
// one thread per (edge, 4-float feature chunk); h/agg live in L2 (25.6MB each)
__global__ void scatter_kernel(const float* __restrict__ h,
                               const int* __restrict__ src,
                               const int* __restrict__ dst,
                               float* __restrict__ agg) {
    int t = blockIdx.x * blockDim.x + threadIdx.x;   // < 25.6M
    int e = t >> 5;          // edge
    int c = t & 31;          // 4-float chunk within 128 feats
    if (e >= NE) return;
    int s = src[e];
    int d = dst[e];
    const float4 v = *(const float4*)(h + (size_t)s * F + c * 4);
    float* o = agg + (size_t)d * F + c * 4;
    atomicAdd(o + 0, v.x);
    atomicAdd(o + 1, v.y);
    atomicAdd(o + 2, v.z);
    atomicAdd(o + 3, v.w);
}

// ------------------------------------------------------- fused SAGE layer GEMM
// out = relu(h @ w_self + (agg*inv_deg) @ w_neigh + b)
// 8 waves/block; each wave computes a 16x128 strip via V_WMMA_F32_16X16X4_F32,
// chaining self+neigh matmuls into the same accumulator.
__global__ void sage_gemm_kernel(const float* __restrict__ h,
                                 const float* __restrict__ agg,
                                 const float* __restrict__ inv_deg,
                                 const float* __restrict__ w_self,
                                 const float* __restrict__ w_neigh,
                                 const float* __restrict__ bias,
                                 float* __restrict__ out) {
    const int lane  = threadIdx.x & 31;
    const int wave  = threadIdx.x >> 5;              // 0..7
    const int row0  = blockIdx.x * 128 + wave * 16;  // first M row of this wave
    const int mrow  = lane & 15;                     // A-fragment row / B col
    const int khalf = (lane >> 4) * 2;               // K sub-offset: 0 or 2

    // clamp loads so EXEC stays all-ones through the WMMAs (ISA requirement)
    const int rowc = min(row0 + mrow, NN - 1);
    const float  idg  = inv_deg[rowc];
    const float* hrow = h   + (size_t)rowc * F;
    const float* arow = agg + (size_t)rowc * F;

    v8f acc[8];
    const v8f zero = {};
#pragma unroll
    for (int n = 0; n < 8; ++n) acc[n] = zero;

    for (int k = 0; k < F; k += 4) {
        // A fragments (16x4 f32): lane half selects K pair, contiguous float2
        v2f a_s, a_n;
        a_s.x = hrow[k + khalf];
        a_s.y = hrow[k + khalf + 1];
        a_n.x = arow[k + khalf] * idg;
        a_n.y = arow[k + khalf + 1] * idg;
#pragma unroll
        for (int n = 0; n < 8; ++n) {
            const int col = n * 16 + mrow;
            // B fragments (4x16 f32): rows k+khalf, k+khalf+1 of the weights
            const float* ws = w_self  + (size_t)(k + khalf) * F + col;
            const float* wn = w_neigh + (size_t)(k + khalf) * F + col;
            v2f b_s, b_n;
            b_s.x = ws[0]; b_s.y = ws[F];
            b_n.x = wn[0]; b_n.y = wn[F];
            acc[n] = __builtin_amdgcn_wmma_f32_16x16x4_f32(
                false, a_s, false, b_s, (short)0, acc[n], false, false);
            acc[n] = __builtin_amdgcn_wmma_f32_16x16x4_f32(
                false, a_n, false, b_n, (short)0, acc[n], false, false);
        }
    }

    // C/D layout: lanes 0-15 -> M = r, lanes 16-31 -> M = 8 + r; N = lane%16
    const int srow = row0 + (lane >> 4) * 8;
#pragma unroll
    for (int n = 0; n < 8; ++n) {
        const int col = n * 16 + mrow;
        const float bv = bias[col];
#pragma unroll
        for (int r = 0; r < 8; ++r) {
            const int orow = srow + r;
            if (orow < NN)
                out[(size_t)orow * F + col] = fmaxf(acc[n][r] + bv, 0.0f);
        }
    }
}

// ---------------------------------------------------------------- mean pool
__global__ void pool_kernel(const float* __restrict__ h, float* __restrict__ accum) {
    const int f   = threadIdx.x & 127;
    const int sub = threadIdx.x >> 7;        // 0..1
    const int base = blockIdx.x * 64;        // 64 nodes per block
    float s = 0.0f;
    for (int i = sub; i < 64; i += 2) {
        const int nidx = base + i;
        if (nidx < NN) s += h[(size_t)nidx * F + f];
    }
    atomicAdd(&accum[f], s);
}

// out[0:16] = score = pls @ w_head + b_head ; out[16:144] = pls = accum / NN
__global__ void head_kernel(const float* __restrict__ accum,
                            const float* __restrict__ w_head,
                            const float* __restrict__ b_head,
                            float* __restrict__ out) {
    const int t = threadIdx.x;               // 128 threads
    const float inv = 1.0f / (float)NN;
    out[16 + t] = accum[t] * inv;
    if (t < NC) {
        float s = b_head[t];
        for (int f = 0; f < F; ++f)
            s += accum[f] * inv * w_head[f * NC + t];
        out[t] = s;
    }
}

// ---------------------------------------------------------------- launcher
extern "C" void kernel_launch(void* const* d_in, const int* in_sizes, int n_in,
                              void* d_out, int out_size, void* d_ws, size_t ws_size,
                              hipStream_t stream) {
    const float* x        = (const float*)d_in[0];
    const int*   src      = (const int*)d_in[1];
    const int*   dst      = (const int*)d_in[2];
    const float* w_self[3]  = { (const float*)d_in[3], (const float*)d_in[6], (const float*)d_in[9]  };
    const float* w_neigh[3] = { (const float*)d_in[4], (const float*)d_in[7], (const float*)d_in[10] };
    const float* bias[3]    = { (const float*)d_in[5], (const float*)d_in[8], (const float*)d_in[11] };
    const float* w_head   = (const float*)d_in[12];
    const float* b_head   = (const float*)d_in[13];
    float* out = (float*)d_out;

    // workspace layout (floats)
    float* ws     = (float*)d_ws;
    float* agg    = ws;                                 // 6.4M
    float* hA     = agg    + (size_t)NN * F;            // 6.4M
    float* hB     = hA     + (size_t)NN * F;            // 6.4M
    float* invdeg = hB     + (size_t)NN * F;            // 50k
    float* accum  = invdeg + NN;                        // 128

    const int feat_elems = NN * F;                      // 6,400,000

    // in-degree -> inverse degree (recomputed each call: deterministic)
    zero_kernel<<<(NN + 255) / 256, 256, 0, stream>>>(invdeg, NN);
    degree_kernel<<<(NE + 255) / 256, 256, 0, stream>>>(dst, invdeg);
    invdeg_kernel<<<(NN + 255) / 256, 256, 0, stream>>>(invdeg);

    const float* hin = x;
    float* houts[3] = { hA, hB, hA };
    for (int l = 0; l < 3; ++l) {
        zero_kernel<<<(feat_elems + 255) / 256, 256, 0, stream>>>(agg, feat_elems);
        scatter_kernel<<<(NE * 32) / 256, 256, 0, stream>>>(hin, src, dst, agg);
        sage_gemm_kernel<<<(NN + 127) / 128, 256, 0, stream>>>(
            hin, agg, invdeg, w_self[l], w_neigh[l], bias[l], houts[l]);
        hin = houts[l];
    }

    zero_kernel<<<1, 128, 0, stream>>>(accum, 128);
    pool_kernel<<<(NN + 63) / 64, 256, 0, stream>>>(hin, accum);
    head_kernel<<<1, 128, 0, stream>>>(accum, w_head, b_head, out);
}